// LegislativeStanceModel_15006615732402
// MI455X (gfx1250) — compile-verified
//
#include <hip/hip_runtime.h>
#include <hip/hip_bf16.h>

#define D_MODEL 128
#define EDGE_DIM 16

typedef float v2f __attribute__((ext_vector_type(2)));
typedef float v8f __attribute__((ext_vector_type(8)));

// D = A(16x4 f32) * B(4x16 f32) + C(16x16 f32), full fp32 precision.
// 8-arg form: (neg_a, A, neg_b, B, c_mod, C, reuse_a, reuse_b)
__device__ __forceinline__ v8f wmma_f32_4(v2f a, v2f b, v8f c) {
    return __builtin_amdgcn_wmma_f32_16x16x4_f32(false, a, false, b, (short)0, c,
                                                 false, false);
}

__device__ __forceinline__ float gelu_exact(float x) {
    return 0.5f * x * (1.0f + erff(x * 0.70710678118654752f));
}

// ---------------------------------------------------------------------------
// Kernel 1: edge gate. Block = 256 thr = 8 waves handles 16 edges.
// Wave w computes h[:,16w:16w+16] = edge_attr(16x16) @ gate_w1[cols]^T via
// 4 x v_wmma_f32_16x16x4_f32, then gelu, dot with gate_w2, sigmoid -> gate[e].
// ---------------------------------------------------------------------------
__global__ void __launch_bounds__(256)
gate_kernel(const float* __restrict__ edge_attr,
            const float* __restrict__ gate_w1, const float* __restrict__ gate_b1,
            const float* __restrict__ gate_w2, const float* __restrict__ gate_b2,
            float* __restrict__ gate, int n_edges) {
    __shared__ float sRow[16];
    const int tid  = threadIdx.x;
    const int wave = tid >> 5;
    const int lane = tid & 31;
    const int e0   = blockIdx.x * 16;

    if (tid < 16) sRow[tid] = 0.0f;
    __syncthreads();

    const int mn   = lane & 15;            // A-row / B-col within tile
    const int koff = (lane >> 4) << 1;     // 0 or 2 (K sub-offset per half-wave)
    const int col  = wave * 16 + mn;       // global hidden dim 0..127
    const int erow = min(e0 + mn, n_edges - 1);  // clamp: keep EXEC all-ones

    const float* aB = edge_attr + (size_t)erow * EDGE_DIM;
    const float* bB = gate_w1   + (size_t)col  * EDGE_DIM;   // gate_w1 is [128,16]

    v8f c = {};
#pragma unroll
    for (int k = 0; k < EDGE_DIM; k += 4) {
        v2f a, b;
        a.x = aB[k + koff]; a.y = aB[k + koff + 1];
        b.x = bB[k + koff]; b.y = bB[k + koff + 1];
        c = wmma_f32_4(a, b, c);
    }

    const float b1 = gate_b1[col];
    const float w2 = gate_w2[col];
#pragma unroll
    for (int j = 0; j < 8; ++j) {
        int m = (lane < 16) ? j : (j + 8);           // D row per ISA layout
        float h = gelu_exact(c[j] + b1) * w2;
        atomicAdd(&sRow[m], h);                      // ds_add_f32
    }
    __syncthreads();

    if (tid < 16) {
        int e = e0 + tid;
        if (e < n_edges) {
            float z = sRow[tid] + gate_b2[0];
            gate[e] = 1.0f / (1.0f + expf(-z));
        }
    }
}

// ---------------------------------------------------------------------------
// Kernel 2: scatter  pre[dst,:] += gate[e] * x_src[src,:],  deg[dst] += 1.
// One wave32 per edge; each lane owns 4 consecutive floats (float4 load).
// Atomics land in L2 (pre = 25.6MB << 192MB L2).
// ---------------------------------------------------------------------------
__global__ void __launch_bounds__(256)
scatter_kernel(const float* __restrict__ x_src,
               const int* __restrict__ edge_src, const int* __restrict__ edge_dst,
               const float* __restrict__ gate,
               float* __restrict__ pre, float* __restrict__ deg, int n_edges) {
    int e    = (blockIdx.x * blockDim.x + threadIdx.x) >> 5;
    int lane = threadIdx.x & 31;
    if (e >= n_edges) return;
    int   src = edge_src[e];
    int   dst = edge_dst[e];
    float g   = gate[e];
    float4 v  = ((const float4*)(x_src + (size_t)src * D_MODEL))[lane];
    float* p  = pre + (size_t)dst * D_MODEL + lane * 4;
    atomicAdd(p + 0, v.x * g);
    atomicAdd(p + 1, v.y * g);
    atomicAdd(p + 2, v.z * g);
    atomicAdd(p + 3, v.w * g);
    if (lane == 0) atomicAdd(deg + dst, 1.0f);
}

// ---------------------------------------------------------------------------
// Kernel 3: fused node update. Block = 256 thr = 8 waves handles 16 nodes.
// c = (pre/deg) @ W_src^T + x_dst @ W_dst^T  (two K=128 WMMA chains into one
// accumulator), + b_dst, spill to LDS, layernorm (half-wave shfl reduce),
// gamma/beta, gelu, store.
// ---------------------------------------------------------------------------
__global__ void __launch_bounds__(256)
node_kernel(const float* __restrict__ pre, const float* __restrict__ deg,
            const float* __restrict__ x_dst,
            const float* __restrict__ W_src, const float* __restrict__ W_dst,
            const float* __restrict__ b_dst,
            const float* __restrict__ ln_gamma, const float* __restrict__ ln_beta,
            float* __restrict__ out, int n_nodes) {
    __shared__ float sTile[16][D_MODEL + 4];
    __shared__ float sInv[16];
    const int tid  = threadIdx.x;
    const int wave = tid >> 5;
    const int lane = tid & 31;
    const int row0 = blockIdx.x * 16;

    if (tid < 16) {
        int r = min(row0 + tid, n_nodes - 1);
        sInv[tid] = 1.0f / fmaxf(deg[r], 1.0f);
    }
    __syncthreads();

    const int mn   = lane & 15;
    const int koff = (lane >> 4) << 1;
    const int col  = wave * 16 + mn;                   // output dim 0..127
    const int arow = min(row0 + mn, n_nodes - 1);      // clamp, EXEC stays full
    const float inv = sInv[mn];

    v8f c = {};
    // GEMM 1: degree-normalized aggregated messages through W_src
    {
        const float* aB = pre   + (size_t)arow * D_MODEL;
        const float* bB = W_src + (size_t)col  * D_MODEL;  // B[k][n] = W[n,k]
        for (int k = 0; k < D_MODEL; k += 4) {
            v2f a, b;
            a.x = aB[k + koff] * inv; a.y = aB[k + koff + 1] * inv;
            b.x = bB[k + koff];       b.y = bB[k + koff + 1];
            c = wmma_f32_4(a, b, c);
        }
    }
    // GEMM 2: residual path x_dst @ W_dst^T into the same accumulator
    {
        const float* aB = x_dst + (size_t)arow * D_MODEL;
        const float* bB = W_dst + (size_t)col  * D_MODEL;
        for (int k = 0; k < D_MODEL; k += 4) {
            v2f a, b;
            a.x = aB[k + koff]; a.y = aB[k + koff + 1];
            b.x = bB[k + koff]; b.y = bB[k + koff + 1];
            c = wmma_f32_4(a, b, c);
        }
    }

    const float bias = b_dst[col];
#pragma unroll
    for (int j = 0; j < 8; ++j) {
        int m = (lane < 16) ? j : (j + 8);
        sTile[m][col] = c[j] + bias;
    }
    __syncthreads();

    // Layernorm: 16 threads (a half-wave) per row, 8 columns per thread.
    const int m  = tid >> 4;
    const int c0 = (tid & 15) * 8;
    float s = 0.0f, ss = 0.0f;
#pragma unroll
    for (int i = 0; i < 8; ++i) {
        float v = sTile[m][c0 + i];
        s += v; ss += v * v;
    }
#pragma unroll
    for (int off = 8; off > 0; off >>= 1) {   // stays within 16-lane group
        s  += __shfl_xor(s,  off, 32);
        ss += __shfl_xor(ss, off, 32);
    }
    const float mean = s * (1.0f / 128.0f);
    const float var  = ss * (1.0f / 128.0f) - mean * mean;
    const float rstd = rsqrtf(var + 1e-5f);

    const int rglob = row0 + m;
    if (rglob < n_nodes) {
#pragma unroll
        for (int i = 0; i < 8; ++i) {
            int cc = c0 + i;
            float v = (sTile[m][cc] - mean) * rstd * ln_gamma[cc] + ln_beta[cc];
            out[(size_t)rglob * D_MODEL + cc] = gelu_exact(v);
        }
    }
}

// ---------------------------------------------------------------------------
extern "C" void kernel_launch(void* const* d_in, const int* in_sizes, int n_in,
                              void* d_out, int out_size, void* d_ws, size_t ws_size,
                              hipStream_t stream) {
    const float* x_src     = (const float*)d_in[0];
    const float* x_dst     = (const float*)d_in[1];
    const int*   edge_src  = (const int*)  d_in[2];
    const int*   edge_dst  = (const int*)  d_in[3];
    const float* edge_attr = (const float*)d_in[4];
    const float* W_src     = (const float*)d_in[5];
    const float* W_dst     = (const float*)d_in[6];
    const float* b_dst     = (const float*)d_in[7];
    const float* gate_w1   = (const float*)d_in[8];
    const float* gate_b1   = (const float*)d_in[9];
    const float* gate_w2   = (const float*)d_in[10];
    const float* gate_b2   = (const float*)d_in[11];
    const float* ln_gamma  = (const float*)d_in[12];
    const float* ln_beta   = (const float*)d_in[13];

    const int n_nodes = in_sizes[0] / D_MODEL;
    const int n_edges = in_sizes[2];

    // Workspace: pre[N*128] | deg[N] | gate[E]
    float* pre  = (float*)d_ws;
    float* deg  = pre + (size_t)n_nodes * D_MODEL;
    float* gate = deg + n_nodes;

    // Zero the atomic accumulators every call (graph-capture safe).
    hipMemsetAsync(d_ws, 0, (size_t)n_nodes * (D_MODEL + 1) * sizeof(float), stream);

    const int gateBlocks = (n_edges + 15) / 16;
    gate_kernel<<<gateBlocks, 256, 0, stream>>>(edge_attr, gate_w1, gate_b1,
                                                gate_w2, gate_b2, gate, n_edges);

    const int scatBlocks = (n_edges + 7) / 8;   // 8 waves/block, 1 edge/wave
    scatter_kernel<<<scatBlocks, 256, 0, stream>>>(x_src, edge_src, edge_dst,
                                                   gate, pre, deg, n_edges);

    const int nodeBlocks = (n_nodes + 15) / 16;
    node_kernel<<<nodeBlocks, 256, 0, stream>>>(pre, deg, x_dst, W_src, W_dst,
                                                b_dst, ln_gamma, ln_beta,
                                                (float*)d_out, n_nodes);
}